// MSSRModule_61383672594883
// MI455X (gfx1250) — compile-verified
//
#include <hip/hip_runtime.h>
#include <hip/hip_bf16.h>
#include <math.h>

// ---------------------------------------------------------------------------
// MSSR (bidirectional Mamba-style SSM) for MI455X / gfx1250.
// GEMMs: V_WMMA_F32_16X16X4_F32, fp32 in / fp32 acc (preserves reference
// numerics; problem is latency-bound: ~3.6 GFLOP, ~27MB, fully L2-resident).
// Each wave computes a 16x64 output strip: 4 independent accumulators share
// one A fragment per K-step. Manual 2-deep software pipeline over K-steps of
// 8 with two explicitly named fragment register sets (S0/S1): each half
// computes on one set, then refills it for K+8, so every fragment load has a
// 4-wmma + 5-load window before its consuming wmma.
// Weights are pre-transposed (417-wide xproj padded to 448) in the workspace
// so ALL fragment loads are maskless contiguous global_load_b64.
// ---------------------------------------------------------------------------

typedef __attribute__((ext_vector_type(2))) float v2f;
typedef __attribute__((ext_vector_type(8))) float v8f;

#define BATCH   2
#define LSEQ    1024          // 32*32
#define DMODEL  192
#define DINNER  384
#define DSTATE  16
#define NXPROJ  417           // 2*16 + 384 + 1 (logical)
#define NXPAD   448           // padded to multiple of 64
#define MROWS   (BATCH*LSEQ)  // 2048

__device__ __forceinline__ float sigmoidf_(float x) { return 1.0f / (1.0f + __expf(-x)); }
__device__ __forceinline__ float siluf_(float x)    { return x * sigmoidf_(x); }
__device__ __forceinline__ float softplusf_(float x){ return (x > 20.0f) ? x : log1pf(expf(x)); }

// ---------------------------------------------------------------------------
// seq[b][l][c] = x[b][c][l]   (NCHW -> NLC)
// ---------------------------------------------------------------------------
__global__ void build_seq_kernel(const float* __restrict__ x, float* __restrict__ seq) {
    int idx = blockIdx.x * blockDim.x + threadIdx.x;
    if (idx >= BATCH * LSEQ * DMODEL) return;
    int c = idx % DMODEL;
    int l = (idx / DMODEL) % LSEQ;
    int b = idx / (DMODEL * LSEQ);
    seq[((size_t)(b * LSEQ + l)) * DMODEL + c] = x[((size_t)(b * DMODEL + c)) * LSEQ + l];
}

// seqf[b][l][c] = seq[b][L-1-l][c]
__global__ void flip_seq_kernel(const float* __restrict__ seq, float* __restrict__ seqf) {
    int idx = blockIdx.x * blockDim.x + threadIdx.x;
    if (idx >= BATCH * LSEQ * DMODEL) return;
    int c = idx % DMODEL;
    int l = (idx / DMODEL) % LSEQ;
    int b = idx / (DMODEL * LSEQ);
    seqf[((size_t)(b * LSEQ + l)) * DMODEL + c] =
        seq[((size_t)(b * LSEQ + (LSEQ - 1 - l))) * DMODEL + c];
}

// ---------------------------------------------------------------------------
// Wt[n][k] = (n < Nsrc) ? W[k][n] : 0    (row-major KxNsrc -> NpadxK, padded)
// ---------------------------------------------------------------------------
__global__ void transpose_w_kernel(const float* __restrict__ W, float* __restrict__ Wt,
                                   int K, int Nsrc, int Npad) {
    int idx = blockIdx.x * blockDim.x + threadIdx.x;
    if (idx >= Npad * K) return;
    int k = idx % K;
    int n = idx / K;
    Wt[(size_t)n * K + k] = (n < Nsrc) ? W[(size_t)k * Nsrc + n] : 0.0f;
}

// ---------------------------------------------------------------------------
// fp32 WMMA GEMM:  C[M x N] = A[M x K] @ Wt^T  (+ optional resid, ld == ldc)
// A row-major MxK; Wt is the weight TRANSPOSED, row-major NxK.
// One wave computes a 16 x (16*NT) strip. N must be a multiple of 16*NT,
// M a multiple of 16, K a multiple of 8 (>= 16). grid = (N/(16*NT), M/16).
//
// Fragments (ISA 7.12.2, 32-bit):
//   A 16x4 : lane t -> row t%16, VGPR j -> K = j + 2*(t/16)
//   B 4x16 : lane t -> col t%16, VGPR j -> K = j + 2*(t/16)
//   D 16x16: VGPR i, lane t -> row i + 8*(t/16), col t%16
// ---------------------------------------------------------------------------
template <int NT>
__global__ __launch_bounds__(32)
void gemm_f32_wmma_kernel(const float* __restrict__ A, const float* __restrict__ Wt,
                          float* __restrict__ C, const float* __restrict__ resid,
                          int M, int K, int N, int ldc) {
    const int lane = threadIdx.x;      // 0..31
    const int r    = lane & 15;
    const int hi   = lane >> 4;
    const int m0   = blockIdx.y * 16;
    const int n0   = blockIdx.x * (16 * NT);

    v8f acc[NT];
    #pragma unroll
    for (int t = 0; t < NT; ++t) acc[t] = (v8f){};

    // this lane's K-walk base pointers (8B-aligned: K mult of 8, offset even)
    const float* Ap = A + (size_t)(m0 + r) * K + 2 * hi;
    const float* Wp[NT];
    #pragma unroll
    for (int t = 0; t < NT; ++t) Wp[t] = Wt + (size_t)(n0 + 16 * t + r) * K + 2 * hi;

    // prologue: fill both fragment sets (k=0 -> S0, k=4 -> S1)
    v2f a0 = *(const v2f*)Ap;
    v2f b0[NT];
    #pragma unroll
    for (int t = 0; t < NT; ++t) b0[t] = *(const v2f*)Wp[t];
    v2f a1 = *(const v2f*)(Ap + 4);
    v2f b1[NT];
    #pragma unroll
    for (int t = 0; t < NT; ++t) b1[t] = *(const v2f*)(Wp[t] + 4);

    for (int k = 8; k < K; k += 8) {
        // compute on S0, then refill S0 from K-step k
        #pragma unroll
        for (int t = 0; t < NT; ++t)
            acc[t] = __builtin_amdgcn_wmma_f32_16x16x4_f32(
                false, a0, false, b0[t], (short)0, acc[t], false, false);
        a0 = *(const v2f*)(Ap + k);
        #pragma unroll
        for (int t = 0; t < NT; ++t) b0[t] = *(const v2f*)(Wp[t] + k);

        // compute on S1, then refill S1 from K-step k+4
        #pragma unroll
        for (int t = 0; t < NT; ++t)
            acc[t] = __builtin_amdgcn_wmma_f32_16x16x4_f32(
                false, a1, false, b1[t], (short)0, acc[t], false, false);
        a1 = *(const v2f*)(Ap + k + 4);
        #pragma unroll
        for (int t = 0; t < NT; ++t) b1[t] = *(const v2f*)(Wp[t] + k + 4);
    }
    // epilogue: last two K-steps
    #pragma unroll
    for (int t = 0; t < NT; ++t)
        acc[t] = __builtin_amdgcn_wmma_f32_16x16x4_f32(
            false, a0, false, b0[t], (short)0, acc[t], false, false);
    #pragma unroll
    for (int t = 0; t < NT; ++t)
        acc[t] = __builtin_amdgcn_wmma_f32_16x16x4_f32(
            false, a1, false, b1[t], (short)0, acc[t], false, false);

    #pragma unroll
    for (int t = 0; t < NT; ++t) {
        const int col = n0 + 16 * t + r;
        #pragma unroll
        for (int i = 0; i < 8; ++i) {
            const int row = m0 + i + 8 * hi;
            float v = acc[t][i];
            size_t idx = (size_t)row * ldc + col;
            if (resid) v += resid[idx];
            C[idx] = v;
        }
    }
}

// ---------------------------------------------------------------------------
// Depthwise causal conv1d (k=4, left pad 3) + bias + SiLU on z = xz[:, :384].
// ---------------------------------------------------------------------------
__global__ void conv_silu_kernel(const float* __restrict__ xz,
                                 const float* __restrict__ conv_w,
                                 const float* __restrict__ conv_b,
                                 float* __restrict__ zact) {
    int idx = blockIdx.x * blockDim.x + threadIdx.x;
    if (idx >= BATCH * LSEQ * DINNER) return;
    int d = idx % DINNER;
    int l = (idx / DINNER) % LSEQ;
    int b = idx / (DINNER * LSEQ);
    float acc = conv_b[d];
    #pragma unroll
    for (int j = 0; j < 4; ++j) {
        int ls = l - 3 + j;
        if (ls >= 0)
            acc += conv_w[d * 4 + j] * xz[((size_t)(b * LSEQ + ls)) * (2 * DINNER) + d];
    }
    zact[((size_t)(b * LSEQ + l)) * DINNER + d] = siluf_(acc);
}

// ---------------------------------------------------------------------------
// Selective scan: one thread per (batch, channel d); 16 states in registers.
// xp row layout (stride NXPAD): [0:16)=B, [16:32)=C, [32:416)=delta, [416]=bias
// ---------------------------------------------------------------------------
__global__ __launch_bounds__(128)
void scan_kernel(const float* __restrict__ xp, const float* __restrict__ zact,
                 const float* __restrict__ xz,        // gate = xz[:, 384:768]
                 const float* __restrict__ A_log, const float* __restrict__ Dp,
                 float* __restrict__ yscan) {
    int idx = blockIdx.x * blockDim.x + threadIdx.x;
    if (idx >= BATCH * DINNER) return;
    int d = idx % DINNER;
    int b = idx / DINNER;

    float Arow[DSTATE];
    #pragma unroll
    for (int s = 0; s < DSTATE; ++s) Arow[s] = -expf(A_log[d * DSTATE + s]);
    const float Dd = Dp[d];

    float h[DSTATE];
    #pragma unroll
    for (int s = 0; s < DSTATE; ++s) h[s] = 0.0f;

    for (int l = 0; l < LSEQ; ++l) {
        size_t row = (size_t)b * LSEQ + l;
        const float* xpr = xp + row * NXPAD;
        float delta = softplusf_(xpr[2 * DSTATE + d] + xpr[NXPROJ - 1]);
        float z = zact[row * DINNER + d];
        float dz = delta * z;
        float acc = 0.0f;
        #pragma unroll
        for (int s = 0; s < DSTATE; ++s) {
            float dA = __expf(delta * Arow[s]);
            h[s] = dA * h[s] + dz * xpr[s];            // dBu = delta*B*z
            acc += h[s] * xpr[DSTATE + s];             // * C
        }
        float y = acc + Dd * z;
        float g = xz[row * (2 * DINNER) + DINNER + d];
        y *= g * sigmoidf_(g);
        yscan[row * DINNER + d] = y;
    }
}

// ---------------------------------------------------------------------------
// LayerNorm over 192 features -> concat buffer (feature offset dir*192),
// rows flipped back for the backward direction.  One wave per row.
// ---------------------------------------------------------------------------
__global__ __launch_bounds__(32)
void layernorm_kernel(const float* __restrict__ y, const float* __restrict__ gamma,
                      const float* __restrict__ beta, float* __restrict__ cat, int dir) {
    int row  = blockIdx.x;
    int lane = threadIdx.x;
    const float* yr = y + (size_t)row * DMODEL;

    float s = 0.0f, s2 = 0.0f;
    for (int c = lane; c < DMODEL; c += 32) {
        float v = yr[c];
        s += v; s2 += v * v;
    }
    #pragma unroll
    for (int off = 16; off > 0; off >>= 1) {
        s  += __shfl_down(s,  off, 32);
        s2 += __shfl_down(s2, off, 32);
    }
    s  = __shfl(s,  0, 32);
    s2 = __shfl(s2, 0, 32);
    float mu   = s / (float)DMODEL;
    float var  = s2 / (float)DMODEL - mu * mu;
    float rstd = rsqrtf(var + 1e-5f);

    int b = row / LSEQ, l = row % LSEQ;
    int drow = b * LSEQ + (dir ? (LSEQ - 1 - l) : l);
    float* out = cat + (size_t)drow * (2 * DMODEL) + dir * DMODEL;
    for (int c = lane; c < DMODEL; c += 32)
        out[c] = (yr[c] - mu) * rstd * gamma[c] + beta[c];
}

// d_out[b][o][l] = tmp[b*L + l][o]   (NLC -> NCHW)
__global__ void store_out_kernel(const float* __restrict__ tmp, float* __restrict__ out) {
    int idx = blockIdx.x * blockDim.x + threadIdx.x;
    if (idx >= BATCH * DMODEL * LSEQ) return;
    int l = idx % LSEQ;
    int o = (idx / LSEQ) % DMODEL;
    int b = idx / (LSEQ * DMODEL);
    out[((size_t)(b * DMODEL + o)) * LSEQ + l] = tmp[((size_t)(b * LSEQ + l)) * DMODEL + o];
}

// ---------------------------------------------------------------------------
extern "C" void kernel_launch(void* const* d_in, const int* in_sizes, int n_in,
                              void* d_out, int out_size, void* d_ws, size_t ws_size,
                              hipStream_t stream) {
    const float* x = (const float*)d_in[0];
    // per-direction params: [in_w, conv_w, conv_b, xproj_w, A_log, D, out_w, ln_g, ln_b]
    const float* P[2][9];
    for (int i = 0; i < 9; ++i) { P[0][i] = (const float*)d_in[1 + i]; P[1][i] = (const float*)d_in[10 + i]; }
    const float* merge_w = (const float*)d_in[19];   // (192, 384) == NxK already
    float* out = (float*)d_out;

    // workspace carve-up (floats) — ~27 MB total
    float* ws    = (float*)d_ws;
    float* seq   = ws;                                    // 2048*192
    float* seqf  = seq   + (size_t)MROWS * DMODEL;        // 2048*192
    float* xz    = seqf  + (size_t)MROWS * DMODEL;        // 2048*768
    float* zact  = xz    + (size_t)MROWS * 2 * DINNER;    // 2048*384
    float* xp    = zact  + (size_t)MROWS * DINNER;        // 2048*448
    float* ysc   = xp    + (size_t)MROWS * NXPAD;         // 2048*384
    float* yout  = ysc   + (size_t)MROWS * DINNER;        // 2048*192
    float* cat   = yout  + (size_t)MROWS * DMODEL;        // 2048*384
    float* wt_in = cat   + (size_t)MROWS * 2 * DMODEL;    // 768*192
    float* wt_xp = wt_in + (size_t)(2 * DINNER) * DMODEL; // 448*384
    float* wt_ou = wt_xp + (size_t)NXPAD * DINNER;        // 192*384
    float* tmp   = wt_ou + (size_t)DMODEL * DINNER;       // 2048*192

    const int TPB  = 256;
    const int nSeq = BATCH * LSEQ * DMODEL;

    build_seq_kernel<<<(nSeq + TPB - 1) / TPB, TPB, 0, stream>>>(x, seq);
    flip_seq_kernel<<<(nSeq + TPB - 1) / TPB, TPB, 0, stream>>>(seq, seqf);

    for (int dir = 0; dir < 2; ++dir) {
        const float* sd = dir ? seqf : seq;
        const float* in_w    = P[dir][0];
        const float* conv_w  = P[dir][1];
        const float* conv_b  = P[dir][2];
        const float* xproj_w = P[dir][3];
        const float* A_log   = P[dir][4];
        const float* Dp      = P[dir][5];
        const float* out_w   = P[dir][6];
        const float* ln_g    = P[dir][7];
        const float* ln_b    = P[dir][8];

        // transpose (and pad) this direction's weights to NxK
        {
            int n;
            n = 2 * DINNER * DMODEL;
            transpose_w_kernel<<<(n + TPB - 1) / TPB, TPB, 0, stream>>>(in_w, wt_in, DMODEL, 2 * DINNER, 2 * DINNER);
            n = NXPAD * DINNER;
            transpose_w_kernel<<<(n + TPB - 1) / TPB, TPB, 0, stream>>>(xproj_w, wt_xp, DINNER, NXPROJ, NXPAD);
            n = DMODEL * DINNER;
            transpose_w_kernel<<<(n + TPB - 1) / TPB, TPB, 0, stream>>>(out_w, wt_ou, DINNER, DMODEL, DMODEL);
        }

        // xz = seq_dir @ in_w     (2048 x 192 x 768)
        {
            dim3 grid((2 * DINNER) / 64, MROWS / 16);
            gemm_f32_wmma_kernel<4><<<grid, 32, 0, stream>>>(sd, wt_in, xz, nullptr,
                                                             MROWS, DMODEL, 2 * DINNER, 2 * DINNER);
        }
        // depthwise conv + SiLU
        {
            int n = BATCH * LSEQ * DINNER;
            conv_silu_kernel<<<(n + TPB - 1) / TPB, TPB, 0, stream>>>(xz, conv_w, conv_b, zact);
        }
        // xp = zact @ xproj_w     (2048 x 384 x 448 padded)
        {
            dim3 grid(NXPAD / 64, MROWS / 16);
            gemm_f32_wmma_kernel<4><<<grid, 32, 0, stream>>>(zact, wt_xp, xp, nullptr,
                                                             MROWS, DINNER, NXPAD, NXPAD);
        }
        // selective scan + D skip + gate
        scan_kernel<<<(BATCH * DINNER + 127) / 128, 128, 0, stream>>>(xp, zact, xz, A_log, Dp, ysc);
        // yout = ysc @ out_w + seq_dir   (2048 x 384 x 192)
        {
            dim3 grid(DMODEL / 64, MROWS / 16);
            gemm_f32_wmma_kernel<4><<<grid, 32, 0, stream>>>(ysc, wt_ou, yout, sd,
                                                             MROWS, DINNER, DMODEL, DMODEL);
        }
        // layernorm -> concat (flip back rows for bwd)
        layernorm_kernel<<<MROWS, 32, 0, stream>>>(yout, ln_g, ln_b, cat, dir);
    }

    // merge: tmp = cat @ merge_w^T    (2048 x 384 x 192); merge_w already NxK
    {
        dim3 grid(DMODEL / 64, MROWS / 16);
        gemm_f32_wmma_kernel<4><<<grid, 32, 0, stream>>>(cat, merge_w, tmp, nullptr,
                                                         MROWS, 2 * DMODEL, DMODEL, DMODEL);
    }
    // NLC -> NCHW
    {
        int n = BATCH * DMODEL * LSEQ;
        store_out_kernel<<<(n + TPB - 1) / TPB, TPB, 0, stream>>>(tmp, out);
    }
}